// yolo_2714419331362
// MI455X (gfx1250) — compile-verified
//
#include <hip/hip_runtime.h>
#include <math.h>

// ---- static problem config (mirrors the reference) ----
#define WGRID   76
#define NA      3
#define NB      16
#define KDIM    85                         // 5 + 80 classes
#define SPATIAL (WGRID*WGRID)              // 5776
#define PER_IMG_BOX (NA*SPATIAL)           // 17328
#define PER_IMG_CH  (PER_IMG_BOX*KDIM)     // 1472880
#define N_DECODED   (NB*PER_IMG_CH)        // 23,566,080
#define N_BOX       (NB*PER_IMG_BOX)       // 277,248
#define M_GT    32
#define TILES_PER_IMG (PER_IMG_BOX/16)     // 1083

typedef float v8f __attribute__((ext_vector_type(8)));
typedef float v2f __attribute__((ext_vector_type(2)));

__device__ __forceinline__ float sigf(float x) { return 1.0f / (1.0f + __builtin_expf(-x)); }
// mask = [0,1,2]: first 3 anchors belong to this head
__device__ __forceinline__ float aw_of(int a) { return a == 0 ? 10.0f : (a == 1 ? 16.0f : 33.0f); }
__device__ __forceinline__ float ah_of(int a) { return a == 0 ? 13.0f : (a == 1 ? 30.0f : 23.0f); }

// ---------------------------------------------------------------------------
// Kernel 1: decode. Pure flat elementwise pass (reshape is layout-identity).
// 4 elements / thread via B128 loads+stores; HBM-bound (~190MB total traffic).
// ---------------------------------------------------------------------------
__global__ void k_decode(const float* __restrict__ pred, float* __restrict__ out) {
    int i4 = (blockIdx.x * blockDim.x + threadIdx.x) * 4;
    if (i4 >= N_DECODED) return;                     // N_DECODED % 4 == 0: all-or-nothing
    const float4 x4 = *reinterpret_cast<const float4*>(pred + i4);
    float xs[4] = {x4.x, x4.y, x4.z, x4.w};
    float rs[4];
#pragma unroll
    for (int u = 0; u < 4; ++u) {
        int idx = i4 + u;
        int k  = idx % KDIM;
        int c  = idx / KDIM;                          // (b,a,w,h) cell index
        int h  = c % WGRID;
        int c2 = c / WGRID;
        int w  = c2 % WGRID;
        int a  = (c2 / WGRID) % NA;
        float x = xs[u];
        float v;
        if (k == 0)      v = (float)h + sigf(x);                      // predx = sig + arange(W) on last axis
        else if (k == 1) v = (float)w + sigf(x);                      // predy = sig + arange(H) on axis-2
        else if (k == 2) v = __builtin_expf(x) * aw_of(a) * 0.125f;   // *AW/608*76, 76/608 == 0.125 exact
        else if (k == 3) v = __builtin_expf(x) * ah_of(a) * 0.125f;
        else             v = sigf(x);                                 // conf + classes
        rs[u] = v;
    }
    *reinterpret_cast<float4*>(out + i4) = make_float4(rs[0], rs[1], rs[2], rs[3]);
}

// ---------------------------------------------------------------------------
// Kernel 2: noobj = (max_m IoU(gt_m, predbox) <= 0.5), plus zero-init of obj.
// One wave per 16-box tile (EXEC all-1s, grid divides exactly). The IoU
// denominator matrix den[i][j] = areaP_i + areaG_j for the 16x16 (box x GT)
// tile is computed with V_WMMA_F32_16X16X4_F32 as a rank-2 outer product:
//   A(16x4) row i = [areaP_i, 1, 0, 0] ; B(4x16) = [1...; areaG_j...; 0; 0]
// D lands in the documented C/D layout: VGPR r -> rows (r, r+8) across lane
// halves, N = lane&15 -> exactly the (box i = r+8*hi, GT j = lane&15) pairing
// used by the VALU min/max intersection below. f32 WMMA is required: pixel
// areas reach ~4e5 which overflows f16.
// ---------------------------------------------------------------------------
__global__ void k_noobj(const float* __restrict__ pred, const float* __restrict__ gt,
                        float* __restrict__ noobj, float* __restrict__ obj) {
    const int lane = threadIdx.x & 31;
    const int wv   = blockIdx.x * (blockDim.x >> 5) + (threadIdx.x >> 5);
    const int img  = wv / TILES_PER_IMG;
    const int base = (wv % TILES_PER_IMG) * 16;      // in-image box tile base
    const int lo   = lane & 15;
    const int hi   = lane >> 4;

    // --- this lane's pred box (box base+lo) in pixel xyxy ---
    const int n   = base + lo;
    const int a   = n / SPATIAL;
    const int rem = n % SPATIAL;
    const int w   = rem / WGRID;
    const int h   = rem % WGRID;
    const float* pp = pred + (long)img * PER_IMG_CH + (long)n * KDIM;
    float t0 = pp[0], t1 = pp[1], t2 = pp[2], t3 = pp[3];
    float px = (float)h + sigf(t0);
    float py = (float)w + sigf(t1);
    float pw = __builtin_expf(t2) * aw_of(a) * 0.125f;
    float ph = __builtin_expf(t3) * ah_of(a) * 0.125f;
    float x1 = (px - 0.5f * pw) * 8.0f, x2 = (px + 0.5f * pw) * 8.0f;   // 608/76 == 8 exact
    float y1 = (py - 0.5f * ph) * 8.0f, y2 = (py + 0.5f * ph) * 8.0f;
    float areaP = (x2 - x1) * (y2 - y1);

    // --- this lane's two GT columns: j = lo (tile0) and j = 16+lo (tile1) ---
    const float* g0 = gt + (long)(img * M_GT + lo) * 6;
    const float* g1 = g0 + 16 * 6;
    float gx1_0 = (g0[1] - 0.5f * g0[3]) * 608.0f, gx2_0 = (g0[1] + 0.5f * g0[3]) * 608.0f;
    float gy1_0 = (g0[2] - 0.5f * g0[4]) * 608.0f, gy2_0 = (g0[2] + 0.5f * g0[4]) * 608.0f;
    float gx1_1 = (g1[1] - 0.5f * g1[3]) * 608.0f, gx2_1 = (g1[1] + 0.5f * g1[3]) * 608.0f;
    float gy1_1 = (g1[2] - 0.5f * g1[4]) * 608.0f, gy2_1 = (g1[2] + 0.5f * g1[4]) * 608.0f;
    float areaG0 = (gx2_0 - gx1_0) * (gy2_0 - gy1_0);
    float areaG1 = (gx2_1 - gx1_1) * (gy2_1 - gy1_1);

    // --- WMMA denominator matrices (one per 16-GT tile) ---
    // A layout (16x4 f32): lanes 0-15 hold K=0 (vgpr0), K=1 (vgpr1); lanes 16-31 hold K=2,3 -> zero.
    const bool lh = lane < 16;
    v2f amat = { lh ? areaP : 0.0f, lh ? 1.0f : 0.0f };
    v2f b0   = { lh ? 1.0f  : 0.0f, lh ? areaG0 : 0.0f };   // rows: K0=ones, K1=areaG, K2=K3=0
    v2f b1   = { lh ? 1.0f  : 0.0f, lh ? areaG1 : 0.0f };
    v8f zero = {};
    v8f den0 = __builtin_amdgcn_wmma_f32_16x16x4_f32(false, amat, false, b0, (short)0, zero, false, false);
    v8f den1 = __builtin_amdgcn_wmma_f32_16x16x4_f32(false, amat, false, b1, (short)0, zero, false, false);

    // --- per-pair intersection + IoU; lane covers (box r+8*hi, GT lo / 16+lo) ---
    float m[8];
#pragma unroll
    for (int r = 0; r < 8; ++r) {
        int src = hi * 8 + r;                         // lane holding box i = r + 8*hi
        float bx1 = __shfl(x1, src), bx2 = __shfl(x2, src);
        float by1 = __shfl(y1, src), by2 = __shfl(y2, src);
        float iw0 = fmaxf(0.0f, fminf(bx2, gx2_0) - fmaxf(bx1, gx1_0));
        float ih0 = fmaxf(0.0f, fminf(by2, gy2_0) - fmaxf(by1, gy1_0));
        float in0 = iw0 * ih0;
        float iou0 = in0 / (den0[r] - in0);           // den from WMMA: areaP_i + areaG_j
        float iw1 = fmaxf(0.0f, fminf(bx2, gx2_1) - fmaxf(bx1, gx1_1));
        float ih1 = fmaxf(0.0f, fminf(by2, gy2_1) - fmaxf(by1, gy1_1));
        float in1 = iw1 * ih1;
        float iou1 = in1 / (den1[r] - in1);
        m[r] = fmaxf(iou0, iou1);
    }
    // max over the 16 GT columns within each lane-half (xor masks stay in-half)
#pragma unroll
    for (int off = 1; off < 16; off <<= 1) {
#pragma unroll
        for (int r = 0; r < 8; ++r) m[r] = fmaxf(m[r], __shfl_xor(m[r], off));
    }

    float* nb = noobj + (long)img * PER_IMG_BOX + base;
    float* ob = obj   + (long)img * PER_IMG_BOX + base;
    if (lo == 0) {
#pragma unroll
        for (int r = 0; r < 8; ++r)
            nb[hi * 8 + r] = (m[r] <= 0.5f) ? 1.0f : 0.0f;
    }
    if (lane < 16) ob[lane] = 0.0f;                   // zero-init obj before the scatter kernel
}

// ---------------------------------------------------------------------------
// Kernel 3: per-GT best-anchor assignment + scatter (runs after k_noobj on
// the same stream). Only constants (0/1) are scattered, so duplicate-index
// races are exact: obj.at[flat].max(valid) and noobj.at[flat].min(~valid).
// ---------------------------------------------------------------------------
__global__ void k_assign(const float* __restrict__ gt,
                         float* __restrict__ noobj, float* __restrict__ obj) {
    int t = blockIdx.x * blockDim.x + threadIdx.x;
    if (t >= NB * M_GT) return;
    const float AW[9] = {10,16,33,30,62,59,116,156,373};
    const float AH[9] = {13,30,23,61,45,119,90,198,326};
    const float* row = gt + (long)t * 6;
    float cx = row[1], cy = row[2], gw = row[3], gh = row[4];
    int b = t / M_GT;
    float gx1 = (cx - 0.5f * gw) * 608.0f, gx2 = (cx + 0.5f * gw) * 608.0f;
    float gy1 = (cy - 0.5f * gh) * 608.0f, gy2 = (cy + 0.5f * gh) * 608.0f;
    float areaG = (gx2 - gx1) * (gy2 - gy1);
    int bi = 0; float bv = -1.0f;
#pragma unroll
    for (int j = 0; j < 9; ++j) {                     // all anchors centered at GT center
        float ow = AW[j] / 608.0f, oh = AH[j] / 608.0f;   // ori = ANCHORS/608 (match ref rounding)
        float ax1 = (cx - 0.5f * ow) * 608.0f, ax2 = (cx + 0.5f * ow) * 608.0f;
        float ay1 = (cy - 0.5f * oh) * 608.0f, ay2 = (cy + 0.5f * oh) * 608.0f;
        float iw = fmaxf(0.0f, fminf(gx2, ax2) - fmaxf(gx1, ax1));
        float ih = fmaxf(0.0f, fminf(gy2, ay2) - fmaxf(gy1, ay1));
        float inter = iw * ih;
        float areaA = (ax2 - ax1) * (ay2 - ay1);
        float iou = inter / (areaG + areaA - inter);
        if (iou > bv) { bv = iou; bi = j; }           // strict '>' == first-argmax
    }
    bool valid = bi < NA;
    int ai = bi < 2 ? bi : 2;                         // clip(best, 0, 2)
    int gi = (int)(cx * (float)WGRID);                // x indexes dim-2 (H) in flat
    int gj = (int)(cy * (float)WGRID);
    int flat = ((b * NA + ai) * WGRID + gi) * WGRID + gj;
    if (valid) { noobj[flat] = 0.0f; obj[flat] = 1.0f; }
}

extern "C" void kernel_launch(void* const* d_in, const int* in_sizes, int n_in,
                              void* d_out, int out_size, void* d_ws, size_t ws_size,
                              hipStream_t stream) {
    (void)in_sizes; (void)n_in; (void)out_size; (void)d_ws; (void)ws_size;
    const float* pred = (const float*)d_in[0];
    const float* gt   = (const float*)d_in[1];
    float* decoded = (float*)d_out;
    float* noobj   = decoded + N_DECODED;   // outputs concatenated flat in return order
    float* obj     = noobj   + N_BOX;

    const int thr = 256;
    int nb1 = (N_DECODED / 4 + thr - 1) / thr;                 // 23014 blocks
    k_decode<<<nb1, thr, 0, stream>>>(pred, decoded);

    int nb2 = (N_BOX / 16) / (thr / 32);                        // 17328 waves / 8 = 2166 blocks (exact)
    k_noobj<<<nb2, thr, 0, stream>>>(pred, gt, noobj, obj);

    k_assign<<<(NB * M_GT + thr - 1) / thr, thr, 0, stream>>>(gt, noobj, obj);
}